// WaveTransformer_23012434772037
// MI455X (gfx1250) — compile-verified
//
#include <hip/hip_runtime.h>
#include <math.h>

// ---------------------------------------------------------------------------
// WaveTransformer fused solve for MI455X (gfx1250, wave32, WMMA).
//
// Only seq position 2 feeds the logits, and the FNet FFT mixing is a fixed
// linear map of the three embedded rows, each of which takes only 113/113/1
// values. Everything up to the MLP nonlinearity is precomputed into small
// L2-resident tables:
//   T0[a], T1[b] in R^4096  (W_in applied to the FFT-mixed embeddings)
//   L0[a], L1[b] in R^113   (direct residual path through unembed)
//   Wu = unembed @ W_out    (113 x 4096)
// Per batch item: h = relu(T0[a]+T1[b]+tc);  logits = Wu h + L0[a]+L1[b]+lc.
// All GEMMs use V_WMMA_F32_16X16X4_F32 for exact fp32 semantics.
//
// Workspace assumption: ws_size >= ~8 MB (uses ~7.7 MB of floats).
// ---------------------------------------------------------------------------

typedef __attribute__((ext_vector_type(2))) float v2f;
typedef __attribute__((ext_vector_type(8))) float v8f;

#define PRIME 113
#define DM    1024
#define DMLP  4096
#define NBATCH 16384
#define NROWS 227      // 113 (a-rows) + 113 (b-rows) + 1 (eq-row)
#define ZROWS 240      // padded to 15 M-tiles of 16
#define NPAD  128      // logit columns padded to 8 N-tiles
#define KCH   512      // K-chunk staged in LDS in the main kernel
#define HS    516      // LDS row stride (516 mod 64 = 4 -> conflict-free tiles)

// --------------------------------------------------------------------------
// Kernel 1: Zall[t] = FFT-mixed (position-2) contribution of token-row t.
//   t in [0,113):    row0 candidates:  C(u)                (cos(0)=1, sin=0)
//   t in [113,226):  row1 candidates: -1/2 C(u) + sqrt3/2 S(u)
//   t == 226:        eq row:  u - 1/2 C(u) - sqrt3/2 S(u)  (incl. identity)
//   t in [227,240):  zero padding
// C/S are the length-1024 cos/sin DFTs; basis built from an LDS table.
// --------------------------------------------------------------------------
__global__ void kZ(const float* __restrict__ te, const float* __restrict__ pe,
                   float* __restrict__ Zall) {
  __shared__ float u[DM], cT[DM], sT[DM];
  const int t = blockIdx.x, tid = threadIdx.x;
  for (int d = tid; d < DM; d += 256) {
    float uv = 0.0f;
    if (t < PRIME)            uv = te[t * DM + d]           + pe[d];
    else if (t < 2 * PRIME)   uv = te[(t - PRIME) * DM + d] + pe[DM + d];
    else if (t == 2 * PRIME)  uv = te[PRIME * DM + d]       + pe[2 * DM + d];
    u[d] = uv;
    float s, c;
    __sincosf(6.283185307179586f * (float)d * (1.0f / 1024.0f), &s, &c);
    cT[d] = c; sT[d] = s;
  }
  __syncthreads();
  if (t > 2 * PRIME) {            // zero padding rows
    for (int i = 0; i < 4; ++i) Zall[(size_t)t * DM + tid + 256 * i] = 0.0f;
    return;
  }
  float ac[4] = {0.f, 0.f, 0.f, 0.f}, as[4] = {0.f, 0.f, 0.f, 0.f};
  const int d0 = tid, d1 = tid + 256, d2 = tid + 512, d3 = tid + 768;
  for (int dp = 0; dp < DM; ++dp) {
    const float uv = u[dp];
    ac[0] += uv * cT[(d0 * dp) & 1023]; as[0] += uv * sT[(d0 * dp) & 1023];
    ac[1] += uv * cT[(d1 * dp) & 1023]; as[1] += uv * sT[(d1 * dp) & 1023];
    ac[2] += uv * cT[(d2 * dp) & 1023]; as[2] += uv * sT[(d2 * dp) & 1023];
    ac[3] += uv * cT[(d3 * dp) & 1023]; as[3] += uv * sT[(d3 * dp) & 1023];
  }
  const float Cc = -0.5f, Ss = 0.8660254037844386f; // cos/sin(2*pi/3) parts
  #pragma unroll
  for (int i = 0; i < 4; ++i) {
    const int d = tid + 256 * i;
    float z;
    if (t < PRIME)          z = ac[i];                          // s'=0
    else if (t < 2 * PRIME) z = Cc * ac[i] + Ss * as[i];        // s'=1
    else                    z = u[d] + Cc * ac[i] - Ss * as[i]; // s'=2 + id
    Zall[(size_t)t * DM + d] = z;
  }
}

// --------------------------------------------------------------------------
// Kernel 1b: zero-pad unembed (113 x 1024) into Upad (128 x 1024) so the
// WMMA GEMM needs no per-lane load guards (branch-free inner loop).
// --------------------------------------------------------------------------
__global__ void kPad(const float* __restrict__ ue, float* __restrict__ Upad) {
  const int idx = blockIdx.x * 256 + threadIdx.x;   // 0 .. 128*1024-1
  const int row = idx >> 10;
  Upad[idx] = (row < PRIME) ? ue[idx] : 0.0f;
}

// --------------------------------------------------------------------------
// Kernel 2: branch-free fp32 WMMA GEMM.  C[m][n] = sum_k A[m][k] * B[k][n].
// A must be zero-padded to a multiple of 16 rows.
//   BKN=1: B stored K-major (K x N row-major, ldb = row stride)
//   BKN=0: B stored N-major (N x K row-major, ldb = row stride)
// One wave per 16x16 tile, 4 N-tiles per 128-thread block, K in steps of 4.
// --------------------------------------------------------------------------
template <int BKN>
__global__ void kGemm(const float* __restrict__ A, const float* __restrict__ B,
                      float* __restrict__ C, int Mstore, int K,
                      int lda, int ldb, int ldc) {
  const int lane  = threadIdx.x & 31;
  const int tileN = blockIdx.x * 4 + (threadIdx.x >> 5);
  const int tileM = blockIdx.y;
  const int r  = lane & 15;      // A-row / B-col / D-col within tile
  const int kh = lane >> 4;      // K half-select
  const float* __restrict__ arow = A + (size_t)(tileM * 16 + r) * lda + 2 * kh;
  const float* __restrict__ bptr =
      BKN ? (B + (size_t)(2 * kh) * ldb + tileN * 16 + r)
          : (B + (size_t)(tileN * 16 + r) * ldb + 2 * kh);
  v8f acc = {};
  #pragma unroll 4
  for (int k = 0; k < K; k += 4) {
    const v2f a = *(const v2f*)(arow + k);
    v2f b;
    if (BKN) { b.x = bptr[(size_t)k * ldb]; b.y = bptr[(size_t)(k + 1) * ldb]; }
    else     { b = *(const v2f*)(bptr + k); }
    acc = __builtin_amdgcn_wmma_f32_16x16x4_f32(false, a, false, b,
                                                (short)0, acc, false, false);
  }
  #pragma unroll
  for (int v = 0; v < 8; ++v) {
    const int mo = tileM * 16 + v + 8 * kh;
    if (mo < Mstore) C[(size_t)mo * ldc + tileN * 16 + r] = acc[v];
  }
}

// --------------------------------------------------------------------------
// Kernel 3: direct-path logit tables. Lall[t][n] = Zall[t] . unembed[n].
// Block 227 builds the constant row: (Zall[226] + b_out) . unembed[n].
// --------------------------------------------------------------------------
__global__ void kL(const float* __restrict__ Zall, const float* __restrict__ ue,
                   const float* __restrict__ bout, float* __restrict__ Lall) {
  __shared__ float zrow[DM];
  const int t = blockIdx.x, tid = threadIdx.x; // 128 threads
  for (int d = tid; d < DM; d += 128)
    zrow[d] = (t < NROWS) ? Zall[(size_t)t * DM + d]
                          : (Zall[(size_t)(2 * PRIME) * DM + d] + bout[d]);
  __syncthreads();
  if (tid < PRIME) {
    float acc = 0.0f;
    for (int d = 0; d < DM; ++d) acc += zrow[d] * ue[(size_t)tid * DM + d];
    Lall[(size_t)t * NPAD + tid] = acc;
  }
}

// --------------------------------------------------------------------------
// Kernel 4: fused main pass. 16 batch rows per block; 8 waves own the 8
// padded 16-col logit tiles. Per 512-wide K chunk: all 256 threads build
// h = relu(T0[a]+T1[b]+Tc+b_in) into LDS, then each wave streams its Wu
// slice from L2 (prefetched one chunk ahead) and issues 128 back-to-back
// v_wmma_f32_16x16x4_f32.
// --------------------------------------------------------------------------
__global__ void __launch_bounds__(256)
kMain(const int* __restrict__ ia, const int* __restrict__ ib,
      const float* __restrict__ Tall, const float* __restrict__ Wu,
      const float* __restrict__ Lall, const float* __restrict__ bin,
      float* __restrict__ out) {
  __shared__ float hbuf[16 * HS];
  __shared__ int   sA[16], sB[16];
  const int tid = threadIdx.x, blk = blockIdx.x;
  if (tid < 16) {
    sA[tid] = ia[blk * 16 + tid];           // Tall/Lall row of a-token
    sB[tid] = ib[blk * 16 + tid] + PRIME;   // Tall/Lall row of b-token
  }
  __syncthreads();
  const int lane = tid & 31, wv = tid >> 5;
  const int r16 = lane & 15, kh = lane >> 4;
  const int ncol = wv * 16 + r16;                         // padded logit col
  const float* __restrict__ wurow = Wu + (size_t)ncol * DMLP;
  const float* __restrict__ Tc    = Tall + (size_t)(2 * PRIME) * DMLP;
  v8f acc = {};
  for (int kb = 0; kb < DMLP; kb += KCH) {
    #pragma unroll
    for (int kt = tid; kt < KCH; kt += 256) {
      const float tc = Tc[kb + kt] + bin[kb + kt];
      #pragma unroll 4
      for (int r = 0; r < 16; ++r) {
        const float hv = Tall[(size_t)sA[r] * DMLP + kb + kt]
                       + Tall[(size_t)sB[r] * DMLP + kb + kt] + tc;
        hbuf[r * HS + kt] = fmaxf(hv, 0.0f);
      }
    }
    __syncthreads();
    if (kb + KCH < DMLP) __builtin_prefetch(wurow + kb + KCH, 0, 0);
    #pragma unroll 8
    for (int kk = 0; kk < KCH; kk += 4) {
      const v2f a = *(const v2f*)(&hbuf[r16 * HS + kk + 2 * kh]);
      const v2f b = *(const v2f*)(wurow + kb + kk + 2 * kh);
      acc = __builtin_amdgcn_wmma_f32_16x16x4_f32(false, a, false, b,
                                                  (short)0, acc, false, false);
    }
    __syncthreads();
  }
  if (ncol < PRIME) {
    const float* __restrict__ Lc = Lall + (size_t)NROWS * NPAD; // const row 227
    #pragma unroll
    for (int v = 0; v < 8; ++v) {
      const int ml = v + 8 * kh;
      const float lg = acc[v]
                     + Lall[(size_t)sA[ml] * NPAD + ncol]
                     + Lall[(size_t)sB[ml] * NPAD + ncol]
                     + Lc[ncol];
      out[(size_t)(blk * 16 + ml) * PRIME + ncol] = lg;
    }
  }
}

// --------------------------------------------------------------------------
extern "C" void kernel_launch(void* const* d_in, const int* in_sizes, int n_in,
                              void* d_out, int out_size, void* d_ws, size_t ws_size,
                              hipStream_t stream) {
  const int*   a    = (const int*)d_in[0];
  const int*   b    = (const int*)d_in[1];
  const float* te   = (const float*)d_in[2];   // (114, 1024)
  const float* pe   = (const float*)d_in[3];   // (3, 1024)
  const float* Win  = (const float*)d_in[4];   // (4096, 1024)
  const float* bin  = (const float*)d_in[5];   // (4096,)
  const float* Wout = (const float*)d_in[6];   // (1024, 4096)
  const float* bout = (const float*)d_in[7];   // (1024,)
  const float* ue   = (const float*)d_in[8];   // (113, 1024)
  float* out = (float*)d_out;

  float* Zall = (float*)d_ws;                  // 240*1024
  float* Tall = Zall + (size_t)ZROWS * DM;     // 240*4096
  float* Wu   = Tall + (size_t)ZROWS * DMLP;   // 128*4096 (rows>=113 unused)
  float* Lall = Wu   + (size_t)NPAD * DMLP;    // 240*128 (rows 0..227 used)
  float* Upad = Lall + (size_t)ZROWS * NPAD;   // 128*1024 zero-padded unembed

  // 1) FFT-mixed token rows + zero-padded unembed
  kZ<<<ZROWS, 256, 0, stream>>>(te, pe, Zall);
  kPad<<<(NPAD * DM) / 256, 256, 0, stream>>>(ue, Upad);
  // 2) Tall = Zall @ W_in^T   (B is N-major: W_in is (4096,1024) row-major)
  {
    dim3 g(DMLP / 64, ZROWS / 16);
    kGemm<0><<<g, 128, 0, stream>>>(Zall, Win, Tall, ZROWS, DM, DM, DM, DMLP);
  }
  // 3) Wu = unembed @ W_out   (B is K-major: W_out is (1024,4096) row-major)
  {
    dim3 g(DMLP / 64, NPAD / 16);
    kGemm<1><<<g, 128, 0, stream>>>(Upad, Wout, Wu, PRIME, DM, DM, DMLP, DMLP);
  }
  // 4) direct-path logit tables (+ const row with b_out)
  kL<<<NROWS + 1, 128, 0, stream>>>(Zall, ue, bout, Lall);
  // 5) fused per-batch pass
  kMain<<<NBATCH / 16, 256, 0, stream>>>(a, b, Tall, Wu, Lall, bin, out);
}